// StepBlur2d_53772990546164
// MI455X (gfx1250) — compile-verified
//
#include <hip/hip_runtime.h>
#include <stdint.h>

// ---- problem geometry -------------------------------------------------------
#define IMG_H 224
#define IMG_W 224
#define N_PLANES (16 * 64)          // batch * channels, each an independent plane

#define TILE_H 16
#define TILE_W 56                   // 224 = 4 * 56
#define HALO   3
#define LDS_H  (TILE_H + 2 * HALO)  // 22
#define LDS_W  (TILE_W + 2 * HALO)  // 62
#define LDS_STRIDE 64               // pad to 64 words: conflict-free, cheap addr math
#define NTHREADS (TILE_W * 4)       // 224 threads = 7 wave32; each thread does 4 rows
#define NELEM (LDS_H * LDS_W)       // 1364 staged elements per block

typedef __attribute__((address_space(1))) int as1_int;   // global
typedef __attribute__((address_space(3))) int as3_int;   // LDS

// CDNA5 async copy: global -> LDS, tracked by ASYNCcnt (no VGPR round trip).
__device__ __forceinline__ void async_copy_b32(const float* gsrc, const float* lds_dst) {
#if __has_builtin(__builtin_amdgcn_global_load_async_to_lds_b32)
  __builtin_amdgcn_global_load_async_to_lds_b32(
      (as1_int*)(uintptr_t)gsrc,
      (as3_int*)(uintptr_t)(uint32_t)(uintptr_t)lds_dst,  // low 32 bits = LDS byte addr
      0, 0);
#else
  uint32_t laddr = (uint32_t)(uintptr_t)lds_dst;
  uint64_t gaddr = (uint64_t)(uintptr_t)gsrc;
  asm volatile("global_load_async_to_lds_b32 %0, %1, off"
               :: "v"(laddr), "v"(gaddr) : "memory");
#endif
}

__device__ __forceinline__ void wait_async_zero() {
#if __has_builtin(__builtin_amdgcn_s_wait_asynccnt)
  __builtin_amdgcn_s_wait_asynccnt(0);
#else
  asm volatile("s_wait_asynccnt 0x0" ::: "memory");
#endif
}

__global__ __launch_bounds__(NTHREADS) void StepBlur2d_kernel(
    const float* __restrict__ img, const float* __restrict__ kf, float* __restrict__ out)
{
  __shared__ float lds[LDS_H * LDS_STRIDE];

  const int plane = blockIdx.z;
  const int y0 = blockIdx.y * TILE_H;
  const int x0 = blockIdx.x * TILE_W;
  const int tid = threadIdx.x;

  const float* base = img + (size_t)plane * (IMG_H * IMG_W);

  // Ring weights, read from the 7x7 kernel (uniform -> scalar loads):
  // center/plus cells have weight kf[3][3]=1; rings picked at representative cells.
  const float w1 = kf[2 * 7 + 2];  // dist sqrt(2)  -> amp^1
  const float w2 = kf[1 * 7 + 1];  // dist 2sqrt(2) -> amp^2
  const float w3 = kf[0 * 7 + 2];  // dist sqrt(10) -> amp^3
  const float w4 = kf[0 * 7 + 0];  // dist 3sqrt(2) -> amp^4

  // Zero-init LDS so skipped (out-of-image) async lanes leave correct 0-padding.
  for (int i = tid; i < LDS_H * LDS_STRIDE; i += NTHREADS) lds[i] = 0.0f;
  __syncthreads();

  // Stage (TILE+halo) tile into LDS with async copies; each HBM element read once.
  for (int i = tid; i < NELEM; i += NTHREADS) {
    const int r = i / LDS_W;
    const int c = i - r * LDS_W;
    const int gy = y0 + r - HALO;
    const int gx = x0 + c - HALO;
    if (gy >= 0 && gy < IMG_H && gx >= 0 && gx < IMG_W) {
      async_copy_b32(base + (size_t)gy * IMG_W + gx, &lds[r * LDS_STRIDE + c]);
    }
  }
  wait_async_zero();   // my wave's async writes to LDS are done
  __syncthreads();     // everyone's are done

  // Each thread: one column, 4 consecutive rows -> 10x7 register window from LDS.
  const int col = tid % TILE_W;
  const int rb  = (tid / TILE_W) * 4;

  float v[10][7];
#pragma unroll
  for (int i = 0; i < 10; ++i)
#pragma unroll
    for (int j = 0; j < 7; ++j)
      v[i][j] = lds[(rb + i) * LDS_STRIDE + col + j];

  float* outp = out + (size_t)plane * (IMG_H * IMG_W);

#pragma unroll
  for (int k = 0; k < 4; ++k) {
#define VV(dy, dx) v[k + 3 + (dy)][3 + (dx)]
    const float c0 = VV(0, 0);
    // e=0 ring (w=1): center + 4-neighbors
    float m0 = fmaxf(fmaxf(fmaxf(VV(-1, 0), VV(1, 0)), fmaxf(VV(0, -1), VV(0, 1))), c0);
    // e=1 ring: diagonals at 1, axis at 2
    float m1 = fmaxf(fmaxf(fmaxf(VV(-1, -1), VV(-1, 1)), fmaxf(VV(1, -1), VV(1, 1))),
                     fmaxf(fmaxf(VV(-2, 0), VV(2, 0)), fmaxf(VV(0, -2), VV(0, 2))));
    // e=2 ring: (1,2),(2,1),(2,2),(3,0) families
    float m2a = fmaxf(fmaxf(VV(-1, -2), VV(-1, 2)), fmaxf(VV(1, -2), VV(1, 2)));
    float m2b = fmaxf(fmaxf(VV(-2, -1), VV(-2, 1)), fmaxf(VV(2, -1), VV(2, 1)));
    float m2c = fmaxf(fmaxf(VV(-2, -2), VV(-2, 2)), fmaxf(VV(2, -2), VV(2, 2)));
    float m2d = fmaxf(fmaxf(VV(-3, 0), VV(3, 0)), fmaxf(VV(0, -3), VV(0, 3)));
    float m2 = fmaxf(fmaxf(m2a, m2b), fmaxf(m2c, m2d));
    // e=3 ring: (1,3),(3,1),(2,3),(3,2) families
    float m3a = fmaxf(fmaxf(VV(-1, -3), VV(-1, 3)), fmaxf(VV(1, -3), VV(1, 3)));
    float m3b = fmaxf(fmaxf(VV(-3, -1), VV(-3, 1)), fmaxf(VV(3, -1), VV(3, 1)));
    float m3c = fmaxf(fmaxf(VV(-2, -3), VV(-2, 3)), fmaxf(VV(2, -3), VV(2, 3)));
    float m3d = fmaxf(fmaxf(VV(-3, -2), VV(-3, 2)), fmaxf(VV(3, -2), VV(3, 2)));
    float m3 = fmaxf(fmaxf(m3a, m3b), fmaxf(m3c, m3d));
    // e=4 ring: 4 corners
    float m4 = fmaxf(fmaxf(VV(-3, -3), VV(-3, 3)), fmaxf(VV(3, -3), VV(3, 3)));

    float o = fmaxf(m0, fmaxf(fmaxf(w1 * m1, w2 * m2), fmaxf(w3 * m3, w4 * m4)));
    o = fmaxf(o, 0.0f);                       // reference's out starts at zeros
    const float res = (c0 > 0.0f) ? c0 : o;   // where(img > 0, img, out)
    outp[(size_t)(y0 + rb + k) * IMG_W + (x0 + col)] = res;
#undef VV
  }
}

extern "C" void kernel_launch(void* const* d_in, const int* in_sizes, int n_in,
                              void* d_out, int out_size, void* d_ws, size_t ws_size,
                              hipStream_t stream) {
  (void)in_sizes; (void)n_in; (void)d_ws; (void)ws_size; (void)out_size;
  const float* img = (const float*)d_in[0];
  const float* kf  = (const float*)d_in[1];
  float* out = (float*)d_out;

  dim3 grid(IMG_W / TILE_W, IMG_H / TILE_H, N_PLANES);  // 4 x 14 x 1024
  dim3 block(NTHREADS);                                  // 224 = 7 wave32
  StepBlur2d_kernel<<<grid, block, 0, stream>>>(img, kf, out);
}